// myModel_57054345560811
// MI455X (gfx1250) — compile-verified
//
#include <hip/hip_runtime.h>
#include <hip/hip_bf16.h>
#include <math.h>

typedef __attribute__((ext_vector_type(2))) float v2f;
typedef __attribute__((ext_vector_type(8))) float v8f;

#define SELU_ALPHA 1.6732632423543772f
#define SELU_SCALE 1.0507009873554805f

__device__ __forceinline__ float selu_f(float x) {
  return x > 0.0f ? SELU_SCALE * x : SELU_SCALE * SELU_ALPHA * (__expf(x) - 1.0f);
}
__device__ __forceinline__ float sigmoid_f(float x) {
  return 1.0f / (1.0f + __expf(-x));
}

__device__ __forceinline__ v8f v8f_zero() {
  v8f z = {0.f, 0.f, 0.f, 0.f, 0.f, 0.f, 0.f, 0.f};
  return z;
}

// D = A(16x4 f32) * B(4x16 f32) + C(16x16 f32)
// A frag: lane holds A[m][2*kh + v], m = lane&15, kh = lane>>4
// B frag: lane holds B[2*kh + v][n], n = lane&15
// C/D:    lane vgpr v holds C[8*kh + v][n]
__device__ __forceinline__ v8f wmma_f32(v2f a, v2f b, v8f c) {
  return __builtin_amdgcn_wmma_f32_16x16x4_f32(false, a, false, b, (short)0, c,
                                               false, false);
}

__global__ __launch_bounds__(256) void zero_kernel(float* __restrict__ p, int n) {
  int i = blockIdx.x * 256 + threadIdx.x;
  if (i < n) p[i] = 0.0f;
}

// ---------------------------------------------------------------------------
// Message MLP: msg = selu([h[first] | h[second]] @ Wmsg + b); agg[second] += msg
// Two 16-pair tiles per wave (B-fragment reused for 2 WMMAs).
// Wmsg in LDS pair-interleaved along K: each B frag is one aligned b64 word.
//   sW[(k>>1)*160 + n*2 + (k&1)] = W[k][n]   (stride 160 => kh halves hit
//   disjoint 32-bank windows; all frag loads are conflict-free ds_load_b64)
// ---------------------------------------------------------------------------
#define MSG_WAVES 4

__global__ __launch_bounds__(128, 1) void msg_kernel(
    const float* __restrict__ h, const int* __restrict__ first,
    const int* __restrict__ second, const float* __restrict__ Wmsg,
    const float* __restrict__ bmsg, float* __restrict__ agg, int M) {
  __shared__ __align__(16) float sW[64 * 160];             // 40 KB interleaved
  __shared__ __align__(16) float sA[MSG_WAVES][2][16 * 132];  // 2 tiles/wave
  __shared__ int sSec[MSG_WAVES][32];

  const int tid = threadIdx.x;
  const int wave = tid >> 5;
  const int lane = tid & 31;

  for (int i = tid; i < 128 * 64; i += 128) {
    const int k = i >> 6, n = i & 63;
    sW[(k >> 1) * 160 + n * 2 + (k & 1)] = Wmsg[i];
  }

  const int row0 = (blockIdx.x * MSG_WAVES + wave) * 32;
  {
    const int r = lane >> 1;
    const int half = lane & 1;  // 0: h[first] -> cols 0..63, 1: h[second]
#pragma unroll
    for (int sub = 0; sub < 2; ++sub) {
      int grow = row0 + sub * 16 + r;
      if (grow > M - 1) grow = M - 1;
      const int src = half ? second[grow] : first[grow];
      const float4* sp = (const float4*)(h + (size_t)src * 64);
      float4* dp = (float4*)(&sA[wave][sub][r * 132 + half * 64]);
#pragma unroll
      for (int j = 0; j < 16; ++j) dp[j] = sp[j];
    }
    const int gr = row0 + lane;
    sSec[wave][lane] = (gr < M) ? second[gr] : -1;
  }
  __syncthreads();

  const int kh = lane >> 4;
  const int nn = lane & 15;

  v8f acc[2][4];
#pragma unroll
  for (int s = 0; s < 2; ++s)
#pragma unroll
    for (int t = 0; t < 4; ++t) acc[s][t] = v8f_zero();

  const float* aBase0 = &sA[wave][0][(lane & 15) * 132 + 2 * kh];
  const float* aBase1 = &sA[wave][1][(lane & 15) * 132 + 2 * kh];
  const float* wBase = &sW[kh * 160 + 2 * nn];

#pragma unroll 4
  for (int ks = 0; ks < 32; ++ks) {
    v2f a0 = *(const v2f*)(aBase0 + ks * 4);  // ds_load_b64
    v2f a1 = *(const v2f*)(aBase1 + ks * 4);
    const float* wr = wBase + ks * 320;
#pragma unroll
    for (int t = 0; t < 4; ++t) {
      v2f b = *(const v2f*)(wr + t * 32);  // ds_load_b64, aligned pair
      acc[0][t] = wmma_f32(a0, b, acc[0][t]);
      acc[1][t] = wmma_f32(a1, b, acc[1][t]);
    }
  }

#pragma unroll
  for (int s = 0; s < 2; ++s) {
#pragma unroll
    for (int t = 0; t < 4; ++t) {
      const int c = t * 16 + nn;
      const float bb = bmsg[c];
#pragma unroll
      for (int v = 0; v < 8; ++v) {
        const int rr = s * 16 + kh * 8 + v;
        const int sidx = sSec[wave][rr];
        if (sidx >= 0) {
          atomicAdd(&agg[(size_t)sidx * 64 + c], selu_f(acc[s][t][v] + bb));
        }
      }
    }
  }
}

// ---------------------------------------------------------------------------
// GRU (Keras reset_after=True), fused: mx = agg@Wk, mh = h@Rk, gates, update.
// Weights pair-interleaved along K in LDS: B frags are single b64 loads.
//   sWk[(k>>1)*416 + c*2 + (k&1)] = Wk[k][c]   (416 mod 64 == 32)
// ---------------------------------------------------------------------------
#define GRU_WAVES 4

__global__ __launch_bounds__(128, 1) void gru_kernel(
    float* __restrict__ h, const float* __restrict__ agg,
    const float* __restrict__ Wk, const float* __restrict__ Rk,
    const float* __restrict__ bin, const float* __restrict__ brec, int E) {
  __shared__ __align__(16) float sWk[32 * 416];  // 52 KB interleaved
  __shared__ __align__(16) float sRk[32 * 416];
  __shared__ __align__(16) float sA1[GRU_WAVES][16 * 68];
  __shared__ __align__(16) float sA2[GRU_WAVES][16 * 68];

  const int tid = threadIdx.x;
  const int wave = tid >> 5;
  const int lane = tid & 31;

  for (int i = tid; i < 64 * 192; i += 128) {
    const int k = i / 192, c = i - k * 192;
    const int o = (k >> 1) * 416 + c * 2 + (k & 1);
    sWk[o] = Wk[i];
    sRk[o] = Rk[i];
  }

  const int row0 = (blockIdx.x * GRU_WAVES + wave) * 16;
  {
    const int r = lane >> 1, half = lane & 1;
    int grow = row0 + r;
    if (grow > E - 1) grow = E - 1;
    const float4* s1 = (const float4*)(agg + (size_t)grow * 64 + half * 32);
    const float4* s2 = (const float4*)(h + (size_t)grow * 64 + half * 32);
    float4* d1 = (float4*)(&sA1[wave][r * 68 + half * 32]);
    float4* d2 = (float4*)(&sA2[wave][r * 68 + half * 32]);
#pragma unroll
    for (int j = 0; j < 8; ++j) {
      d1[j] = s1[j];
      d2[j] = s2[j];
    }
  }
  __syncthreads();

  const int kh = lane >> 4;
  const int nn = lane & 15;
  const float* a1Base = &sA1[wave][(lane & 15) * 68 + 2 * kh];
  const float* a2Base = &sA2[wave][(lane & 15) * 68 + 2 * kh];
  const float* wkBase = &sWk[kh * 416 + 2 * nn];
  const float* rkBase = &sRk[kh * 416 + 2 * nn];

#pragma unroll
  for (int tt = 0; tt < 4; ++tt) {
    const int cb = tt * 16;
    v8f xz = v8f_zero(), xr = v8f_zero(), xh = v8f_zero();
    v8f hz = v8f_zero(), hr = v8f_zero(), hh = v8f_zero();
#pragma unroll 4
    for (int ks = 0; ks < 16; ++ks) {
      v2f a1 = *(const v2f*)(a1Base + ks * 4);
      v2f a2 = *(const v2f*)(a2Base + ks * 4);
      const float* wk = wkBase + ks * 832;
      const float* rk = rkBase + ks * 832;
      v2f b0 = *(const v2f*)(wk + cb * 2);
      v2f b1 = *(const v2f*)(wk + (64 + cb) * 2);
      v2f b2 = *(const v2f*)(wk + (128 + cb) * 2);
      v2f c0 = *(const v2f*)(rk + cb * 2);
      v2f c1 = *(const v2f*)(rk + (64 + cb) * 2);
      v2f c2 = *(const v2f*)(rk + (128 + cb) * 2);
      xz = wmma_f32(a1, b0, xz);
      xr = wmma_f32(a1, b1, xr);
      xh = wmma_f32(a1, b2, xh);
      hz = wmma_f32(a2, c0, hz);
      hr = wmma_f32(a2, c1, hr);
      hh = wmma_f32(a2, c2, hh);
    }
    const int c = cb + nn;
    const float biz = bin[c], bir = bin[64 + c], bih = bin[128 + c];
    const float brz = brec[c], brr = brec[64 + c], brh = brec[128 + c];
#pragma unroll
    for (int v = 0; v < 8; ++v) {
      const int rr = kh * 8 + v;
      const int grow = row0 + rr;
      if (grow < E) {
        const float z = sigmoid_f(xz[v] + biz + hz[v] + brz);
        const float rg = sigmoid_f(xr[v] + bir + hr[v] + brr);
        const float hc = tanhf(xh[v] + bih + rg * (hh[v] + brh));
        const float ho = sA2[wave][rr * 68 + c];
        h[(size_t)grow * 64 + c] = z * ho + (1.0f - z) * hc;
      }
    }
  }
}

// ---------------------------------------------------------------------------
// Readout
// ---------------------------------------------------------------------------
__global__ __launch_bounds__(256) void graph_sum_kernel(
    const float* __restrict__ h, const int* __restrict__ gids,
    float* __restrict__ g, int E) {
  int i = blockIdx.x * 256 + threadIdx.x;
  if (i < E * 64) {
    const int e = i >> 6, c = i & 63;
    atomicAdd(&g[(size_t)gids[e] * 64 + c], h[i]);
  }
}

template <int KDIM, bool DOSELU>
__global__ __launch_bounds__(256) void fc_kernel(
    const float* __restrict__ A, const float* __restrict__ W,
    const float* __restrict__ bias, float* __restrict__ out, int nRows,
    int nCols) {
  const int wave = (blockIdx.x * blockDim.x + threadIdx.x) >> 5;
  const int lane = threadIdx.x & 31;
  const int tilesC = nCols >> 4;
  const int rt = wave / tilesC;
  const int ct = wave - rt * tilesC;
  if (rt * 16 >= nRows) return;  // wave-uniform
  const int kh = lane >> 4;
  const int nn = lane & 15;
  const int m = rt * 16 + (lane & 15);
  const float* aP = A + (size_t)m * KDIM + 2 * kh;
  const float* wP = W + (size_t)(2 * kh) * nCols + ct * 16 + nn;
  v8f acc = v8f_zero();
#pragma unroll 4
  for (int ks = 0; ks < KDIM / 4; ++ks) {
    v2f a = *(const v2f*)(aP + ks * 4);
    v2f b = {wP[(size_t)ks * 4 * nCols], wP[(size_t)ks * 4 * nCols + nCols]};
    acc = wmma_f32(a, b, acc);
  }
  const float bb = bias[ct * 16 + nn];
#pragma unroll
  for (int v = 0; v < 8; ++v) {
    const int row = rt * 16 + kh * 8 + v;
    if (row < nRows) {
      float x = acc[v] + bb;
      if (DOSELU) x = selu_f(x);
      out[(size_t)row * nCols + ct * 16 + nn] = x;
    }
  }
}

__global__ __launch_bounds__(256) void fc3_kernel(
    const float* __restrict__ r2, const float* __restrict__ W3,
    const float* __restrict__ b3, float* __restrict__ out, int Gn) {
  const int g = blockIdx.x * 8 + (threadIdx.x >> 5);
  const int lane = threadIdx.x & 31;
  if (g >= Gn) return;
  float s = 0.0f;
#pragma unroll
  for (int i = lane; i < 256; i += 32) s += r2[(size_t)g * 256 + i] * W3[i];
#pragma unroll
  for (int o = 16; o > 0; o >>= 1) s += __shfl_down(s, o, 32);
  if (lane == 0) out[g] = s + b3[0];
}

// ---------------------------------------------------------------------------
extern "C" void kernel_launch(void* const* d_in, const int* in_sizes, int n_in,
                              void* d_out, int out_size, void* d_ws,
                              size_t ws_size, hipStream_t stream) {
  (void)n_in;
  (void)ws_size;
  const float* states_action = (const float*)d_in[0];
  const int* gids = (const int*)d_in[1];
  const int* first = (const int*)d_in[2];
  const int* second = (const int*)d_in[3];
  // d_in[4] = num_edges scalar (== E, redundant with in_sizes[1])
  const float* Wmsg = (const float*)d_in[5];
  const float* bmsg = (const float*)d_in[6];
  const float* gk = (const float*)d_in[7];
  const float* grk = (const float*)d_in[8];
  const float* gbi = (const float*)d_in[9];
  const float* gbr = (const float*)d_in[10];
  const float* W1 = (const float*)d_in[11];
  const float* b1 = (const float*)d_in[12];
  const float* W2 = (const float*)d_in[13];
  const float* b2 = (const float*)d_in[14];
  const float* W3 = (const float*)d_in[15];
  const float* b3 = (const float*)d_in[16];

  const int E = in_sizes[1];
  const int M = in_sizes[2];
  const int Gn = out_size;

  float* h = (float*)d_ws;
  float* agg = h + (size_t)E * 64;
  float* g = agg + (size_t)E * 64;
  float* r1 = g + (size_t)Gn * 64;
  float* r2 = r1 + (size_t)Gn * 256;

  hipMemcpyAsync(h, states_action, (size_t)E * 64 * sizeof(float),
                 hipMemcpyDeviceToDevice, stream);

  const int pairTiles = (M + 31) / 32;  // 32 rows per wave (2 WMMA tiles)
  const int msgBlocks = (pairTiles + MSG_WAVES - 1) / MSG_WAVES;
  const int tilesE = (E + 15) / 16;
  const int gruBlocks = (tilesE + GRU_WAVES - 1) / GRU_WAVES;
  const int nAgg = E * 64;

  for (int it = 0; it < 8; ++it) {
    zero_kernel<<<(nAgg + 255) / 256, 256, 0, stream>>>(agg, nAgg);
    msg_kernel<<<msgBlocks, 128, 0, stream>>>(h, first, second, Wmsg, bmsg, agg,
                                              M);
    gru_kernel<<<gruBlocks, 128, 0, stream>>>(h, agg, gk, grk, gbi, gbr, E);
  }

  zero_kernel<<<(Gn * 64 + 255) / 256, 256, 0, stream>>>(g, Gn * 64);
  graph_sum_kernel<<<(nAgg + 255) / 256, 256, 0, stream>>>(h, gids, g, E);

  const int tilesG = (Gn + 15) / 16;
  const int fcWaves = tilesG * (256 / 16);
  fc_kernel<64, true><<<(fcWaves * 32 + 255) / 256, 256, 0, stream>>>(
      g, W1, b1, r1, Gn, 256);
  fc_kernel<256, true><<<(fcWaves * 32 + 255) / 256, 256, 0, stream>>>(
      r1, W2, b2, r2, Gn, 256);
  fc3_kernel<<<(Gn + 7) / 8, 256, 0, stream>>>(r2, W3, b3, (float*)d_out, Gn);
}